// First_n_Last_65678639890655
// MI455X (gfx1250) — compile-verified
//
#include <hip/hip_runtime.h>

typedef float v2f __attribute__((ext_vector_type(2)));
typedef float v8f __attribute__((ext_vector_type(8)));

#define SEQ    1024
#define HID    1024
#define NSPANS 12288
#define KDIM   2048          // 2*HID
#define MTOT   49152         // BATCH * NSPANS

#define BM  64               // span rows per block
#define BN  256              // output cols per block
#define KC  128              // K chunk staged in LDS
#define LDA 130              // padded LDS row stride (floats)

__global__ __launch_bounds__(256)
void First_n_Last_span_gemm(const float* __restrict__ tok,
                            const int*   __restrict__ span_ids,
                            const int*   __restrict__ pooling,
                            const float* __restrict__ W,
                            const float* __restrict__ bias,
                            float* __restrict__ out)
{
    __shared__ float As[BM * LDA];          // 64 x 128 A-chunk, padded
    __shared__ int   sBase[BM];             // b * SEQ  (token row base)
    __shared__ int   sLo0[BM], sHi0[BM];    // start window [lo0, hi0)
    __shared__ int   sLo1[BM], sHi1[BM];    // end   window [lo1, hi1)
    __shared__ int   sValid[BM];            // span_len > 0

    const int tid = threadIdx.x;
    const int M0  = blockIdx.x * BM;
    const int N0  = blockIdx.y * BN;

    // ---- per-span metadata (one thread per span row) ----
    if (tid < BM) {
        const int m   = M0 + tid;
        const int b   = m / NSPANS;
        const int st  = span_ids[2 * m + 0];
        const int en  = span_ids[2 * m + 1];
        const int win = (pooling[0] != 0) ? 1 : 3;
        sBase[tid]  = b * SEQ;
        sLo0[tid]   = st;
        sHi0[tid]   = min(st + win, en);
        sLo1[tid]   = max(en - win, st);
        sHi1[tid]   = en;
        sValid[tid] = (en > st) ? 1 : 0;
    }

    const int lane = tid & 31;
    const int wv   = tid >> 5;         // wave id 0..7
    const int q    = lane & 15;        // column-in-tile / row-in-tile
    const int half = lane >> 4;        // K sub-group select
    const int nw   = N0 + wv * 32;     // this wave's first output column

    v8f acc[4][2];
    #pragma unroll
    for (int i = 0; i < 4; ++i)
        #pragma unroll
        for (int j = 0; j < 2; ++j)
            acc[i][j] = (v8f)0.0f;

    // ---- main loop over K chunks ----
    for (int kc = 0; kc < KDIM; kc += KC) {
        __syncthreads();   // protect As (and metadata on first pass)

        // build A chunk: windowed max gather, coalesced along H
        for (int e = tid; e < BM * KC; e += 256) {
            const int r = e >> 7;        // span row 0..63
            const int c = e & 127;       // col in chunk
            const int k = kc + c;        // global K index
            float v = 0.0f;
            if (sValid[r]) {
                int h, lo, hi;
                if (k < HID) { h = k;       lo = sLo0[r]; hi = sHi0[r]; }
                else         { h = k - HID; lo = sLo1[r]; hi = sHi1[r]; }
                const float* p = tok + (sBase[r] + lo) * HID + h;
                v = p[0];
                if (lo + 1 < hi) v = fmaxf(v, p[HID]);
                if (lo + 2 < hi) v = fmaxf(v, p[2 * HID]);
            }
            As[r * LDA + c] = v;
        }
        __syncthreads();

        const float* Wc = W + (size_t)kc;
        for (int kk = 0; kk < KC; kk += 4) {
            const int kl = kk + 2 * half;
            // A fragments: 4 M-tiles from LDS (f32 16x4 layout)
            v2f a0 = *(const v2f*)&As[(0 * 16 + q) * LDA + kl];
            v2f a1 = *(const v2f*)&As[(1 * 16 + q) * LDA + kl];
            v2f a2 = *(const v2f*)&As[(2 * 16 + q) * LDA + kl];
            v2f a3 = *(const v2f*)&As[(3 * 16 + q) * LDA + kl];
            // B fragments: W^T straight from (L2-resident) W
            v2f b0 = *(const v2f*)&Wc[(size_t)(nw + q)      * KDIM + kl];
            v2f b1 = *(const v2f*)&Wc[(size_t)(nw + 16 + q) * KDIM + kl];

            acc[0][0] = __builtin_amdgcn_wmma_f32_16x16x4_f32(false, a0, false, b0, (short)0, acc[0][0], false, false);
            acc[0][1] = __builtin_amdgcn_wmma_f32_16x16x4_f32(false, a0, false, b1, (short)0, acc[0][1], false, false);
            acc[1][0] = __builtin_amdgcn_wmma_f32_16x16x4_f32(false, a1, false, b0, (short)0, acc[1][0], false, false);
            acc[1][1] = __builtin_amdgcn_wmma_f32_16x16x4_f32(false, a1, false, b1, (short)0, acc[1][1], false, false);
            acc[2][0] = __builtin_amdgcn_wmma_f32_16x16x4_f32(false, a2, false, b0, (short)0, acc[2][0], false, false);
            acc[2][1] = __builtin_amdgcn_wmma_f32_16x16x4_f32(false, a2, false, b1, (short)0, acc[2][1], false, false);
            acc[3][0] = __builtin_amdgcn_wmma_f32_16x16x4_f32(false, a3, false, b0, (short)0, acc[3][0], false, false);
            acc[3][1] = __builtin_amdgcn_wmma_f32_16x16x4_f32(false, a3, false, b1, (short)0, acc[3][1], false, false);
        }
    }

    // ---- epilogue: bias + store (f32 C/D layout: row = j + 8*half, col = q) ----
    const float bias0 = bias[nw + q];
    const float bias1 = bias[nw + 16 + q];
    #pragma unroll
    for (int mt = 0; mt < 4; ++mt) {
        #pragma unroll
        for (int nt = 0; nt < 2; ++nt) {
            const float bv   = nt ? bias1 : bias0;
            const int   ncol = nw + nt * 16 + q;
            #pragma unroll
            for (int j = 0; j < 8; ++j) {
                const int mrow = M0 + mt * 16 + j + 8 * half;
                out[(size_t)mrow * HID + ncol] = acc[mt][nt][j] + bv;
            }
        }
    }
}

extern "C" void kernel_launch(void* const* d_in, const int* in_sizes, int n_in,
                              void* d_out, int out_size, void* d_ws, size_t ws_size,
                              hipStream_t stream) {
    (void)in_sizes; (void)n_in; (void)out_size; (void)d_ws; (void)ws_size;
    const float* tok     = (const float*)d_in[0];   // (4,1024,1024) f32
    const int*   span    = (const int*)  d_in[1];   // (4,12288,2)   int
    const int*   pooling = (const int*)  d_in[2];   // scalar int
    const float* W       = (const float*)d_in[3];   // (1024,2048)   f32
    const float* bias    = (const float*)d_in[4];   // (1024,)       f32
    float*       out     = (float*)d_out;           // (4,12288,1024) f32

    dim3 grid(MTOT / BM, HID / BN);   // (768, 4)
    First_n_Last_span_gemm<<<grid, dim3(256), 0, stream>>>(tok, span, pooling, W, bias, out);
}